// NodeMixingResBlock_67869073211644
// MI455X (gfx1250) — compile-verified
//
#include <hip/hip_runtime.h>

typedef __attribute__((ext_vector_type(2))) float v2f;
typedef __attribute__((ext_vector_type(4))) float v4f;
typedef __attribute__((ext_vector_type(8))) float v8f;

#define B_DIM   16
#define N_NODES 256
#define T_WIN   64
#define F_DIM   128
#define XS_STRIDE 132   // 128 + 4 floats of padding -> bank-conflict-free B reads

// LDS layout: Xs[256][132] f32  |  meta[256] u32  (packed jobid<<16 | entryid)
// total = (256*132 + 256) * 4 = 136192 bytes (< 320 KB WGP limit)

__global__ __launch_bounds__(256, 1)
void node_mixing_wmma_kernel(const float* __restrict__ x,
                             const int*   __restrict__ entryid,
                             const int*   __restrict__ jobid,
                             const float* __restrict__ fw,
                             float*       __restrict__ out)
{
    extern __shared__ float smem[];
    float*    Xs   = smem;
    unsigned* meta = (unsigned*)(smem + N_NODES * XS_STRIDE);

    const int bid = blockIdx.x;
    const int b   = bid >> 6;        // / T_WIN
    const int t   = bid & 63;        // % T_WIN

    const int tid  = threadIdx.x;
    const int lane = tid & 31;
    const int w    = tid >> 5;       // wave id 0..7

    // ---- stage per-k metadata: meta[k] = (jobid << 16) | entryid ----
    {
        const int k   = tid;
        const int idx = (b * N_NODES + k) * T_WIN + t;
        const unsigned e = (unsigned)entryid[idx] & 0xFFFFu;
        const unsigned j = (unsigned)jobid[idx];
        meta[k] = (j << 16) | e;
    }

    // ---- stage X[b,:,t,:] into LDS; one row per wave-iteration (512B coalesced) ----
    #pragma unroll 4
    for (int i = 0; i < 32; ++i) {
        const int row = w * 32 + i;
        const float* src = x + ((size_t)((b * N_NODES + row) * T_WIN + t)) * F_DIM;
        v4f val = __builtin_nontemporal_load((const v4f*)(src) + lane);
        *(v4f*)(Xs + row * XS_STRIDE + lane * 4) = val;
    }
    __syncthreads();

    const int r     = lane & 15;     // row-in-tile / column index
    const int khalf = lane >> 4;     // which K-pair this lane holds

    const int l0 = 32 * w + r;       // A-matrix rows owned by this lane
    const int l1 = l0 + 16;
    const unsigned jl0 = meta[l0] >> 16;
    const unsigned jl1 = meta[l1] >> 16;
    const float* fwrow0 = fw + l0 * N_NODES;
    const float* fwrow1 = fw + l1 * N_NODES;

    v8f acc0[8], acc1[8];
    #pragma unroll
    for (int nt = 0; nt < 8; ++nt) { acc0[nt] = (v8f){}; acc1[nt] = (v8f){}; }

    // ---- K loop: Y[256x128] += W_eff[256x256] * X[256x128], K step 4 ----
    #pragma unroll 2
    for (int kk = 0; kk < N_NODES; kk += 4) {
        const int k0 = kk + 2 * khalf;          // lane's K indices: k0, k0+1
        const unsigned p0 = meta[k0];
        const unsigned p1 = meta[k0 + 1];
        const unsigned e0 = p0 & 0xFFFFu, j0 = p0 >> 16;
        const unsigned e1 = p1 & 0xFFFFu, j1 = p1 >> 16;

        // branch-free job-equality mask (keeps EXEC all-ones for WMMA)
        const float m00 = (j0 == jl0 && j0 != 0u) ? 1.0f : 0.0f;
        const float m01 = (j1 == jl0 && j1 != 0u) ? 1.0f : 0.0f;
        const float m10 = (j0 == jl1 && j0 != 0u) ? 1.0f : 0.0f;
        const float m11 = (j1 == jl1 && j1 != 0u) ? 1.0f : 0.0f;

        v2f A0, A1;                              // gathered + masked A fragments
        A0.x = fwrow0[e0] * m00;  A0.y = fwrow0[e1] * m01;
        A1.x = fwrow1[e0] * m10;  A1.y = fwrow1[e1] * m11;

        const float* xr0 = Xs + k0 * XS_STRIDE + r;
        const float* xr1 = xr0 + XS_STRIDE;

        #pragma unroll
        for (int nt = 0; nt < 8; ++nt) {
            v2f Bf;
            Bf.x = xr0[nt * 16];
            Bf.y = xr1[nt * 16];
            acc0[nt] = __builtin_amdgcn_wmma_f32_16x16x4_f32(
                false, A0, false, Bf, (short)0, acc0[nt], false, false);
            acc1[nt] = __builtin_amdgcn_wmma_f32_16x16x4_f32(
                false, A1, false, Bf, (short)0, acc1[nt], false, false);
        }
    }

    // ---- epilogue: relu + residual (residual re-read from LDS, not HBM) ----
    #pragma unroll
    for (int nt = 0; nt < 8; ++nt) {
        const int f = nt * 16 + r;
        #pragma unroll
        for (int v = 0; v < 8; ++v) {
            const int la = 32 * w + 8 * khalf + v;   // C-layout: M = v + 8*(lane>=16)
            const int lb = la + 16;
            float ya = acc0[nt][v]; ya = ya > 0.0f ? ya : 0.0f;
            float yb = acc1[nt][v]; yb = yb > 0.0f ? yb : 0.0f;
            const float ra = Xs[la * XS_STRIDE + f];
            const float rb = Xs[lb * XS_STRIDE + f];
            float* pa = out + ((size_t)((b * N_NODES + la) * T_WIN + t)) * F_DIM + f;
            float* pb = out + ((size_t)((b * N_NODES + lb) * T_WIN + t)) * F_DIM + f;
            __builtin_nontemporal_store(ra + ya, pa);
            __builtin_nontemporal_store(rb + yb, pb);
        }
    }
}

extern "C" void kernel_launch(void* const* d_in, const int* in_sizes, int n_in,
                              void* d_out, int out_size, void* d_ws, size_t ws_size,
                              hipStream_t stream) {
    const float* x       = (const float*)d_in[0];
    const int*   entryid = (const int*)  d_in[1];
    const int*   jobid   = (const int*)  d_in[2];
    // d_in[3] = rackid (dead), d_in[5..7] = bias/gamma/beta (dead)
    const float* fw      = (const float*)d_in[4];
    float*       out     = (float*)d_out;

    const dim3 grid(B_DIM * T_WIN);   // 1024 workgroups, one per (b, t) GEMM
    const dim3 block(256);            // 8 waves
    const size_t shmem = (size_t)(N_NODES * XS_STRIDE + N_NODES) * sizeof(float);
    node_mixing_wmma_kernel<<<grid, block, shmem, stream>>>(x, entryid, jobid, fw, out);
}